// RAGFormerModel_17403207483542
// MI455X (gfx1250) — compile-verified
//
#include <hip/hip_runtime.h>
#include <hip/hip_bf16.h>
#include <math.h>

// ---------------------------------------------------------------------------
// RAGFormer for MI455X (gfx1250, wave32).  All GEMM-like math uses
// V_WMMA_F32_16X16X4_F32 (full f32 precision, matches the f32 reference).
// GEMMs are register-blocked 16x64 per wave: one A fragment feeds 4
// independent accumulators (4x less A refetch, hazard-free WMMA ILP).
// Workload is activation/L2-bandwidth bound; hist (84MB) is L2-resident.
// ---------------------------------------------------------------------------

#define Bsz   256
#define Tlen  60
#define Dm    128
#define Hn    8
#define HDm   16
#define Lnum  4
#define FFm   512
#define DIN   21
#define Ktop  7
#define NHIST 1000000

typedef __attribute__((ext_vector_type(2))) float v2f;
typedef __attribute__((ext_vector_type(8))) float v8f;

__device__ __forceinline__ float gelu_exact(float x) {
    return 0.5f * x * (1.0f + erff(x * 0.70710678118654752f));
}

__device__ __forceinline__ float wave_sum32(float v) {
    #pragma unroll
    for (int off = 16; off > 0; off >>= 1) v += __shfl_xor(v, off, 32);
    return v;
}

// ---------------------------------------------------------------------------
// Input projection + positional encoding:  h[bt,d] = x[bt,:]@w_in + b_in + PE
// ---------------------------------------------------------------------------
__global__ __launch_bounds__(128) void inproj_kernel(
    const float* __restrict__ x, const float* __restrict__ w_in,
    const float* __restrict__ b_in, float* __restrict__ h) {
    const int bt = blockIdx.x;          // 0 .. B*T-1
    const int d  = threadIdx.x;         // 0 .. 127
    const int t  = bt % Tlen;
    const float* xr = x + (size_t)bt * DIN;
    float acc = b_in[d];
    #pragma unroll
    for (int j = 0; j < DIN; ++j) acc += xr[j] * w_in[j * Dm + d];
    // sinusoidal PE computed on device: div = exp(-(2*(d/2)) * ln(10000)/128)
    const int i2 = d & ~1;
    const float div = __expf(-0.0719557945f * (float)i2);
    const float ang = (float)t * div;
    acc += (d & 1) ? __cosf(ang) : __sinf(ang);
    h[(size_t)bt * Dm + d] = acc;
}

// ---------------------------------------------------------------------------
// f32 WMMA GEMM, 16x64 tile per wave: C[M,N] = A[M,K] @ B[K,N] + bias
// (+resid) (+gelu).  One A fragment is shared by 4 B fragments -> 4
// independent accumulator chains (good WMMA ILP, 4x less A traffic).
// ---------------------------------------------------------------------------
template<bool GELU, bool RESID>
__global__ __launch_bounds__(32) void gemm16x64_f32(
    const float* __restrict__ A, const float* __restrict__ B,
    const float* __restrict__ bias, const float* __restrict__ R,
    float* __restrict__ C, int M, int N, int K) {
    const int n0   = blockIdx.x * 64;
    const int m0   = blockIdx.y * 16;
    const int lane = threadIdx.x;       // 0..31, all active (WMMA needs EXEC=all)
    const int lm   = lane & 15;
    const int hi   = lane >> 4;         // 0 or 1

    v8f acc0 = {}, acc1 = {}, acc2 = {}, acc3 = {};
    const float* arow = A + (size_t)(m0 + lm) * K;   // A-frag: row m0+lm
    const float* bp   = B + n0 + lm;                 // B-frag base: col n0+lm
    #pragma unroll 2
    for (int k0 = 0; k0 < K; k0 += 4) {
        v2f a;
        // 32-bit A 16x4 layout: VGPR0 = K={0,2}, VGPR1 = K={1,3} per half-wave
        a[0] = arow[k0 + 2 * hi];
        a[1] = arow[k0 + 2 * hi + 1];
        const float* b0 = bp + (size_t)(k0 + 2 * hi) * N;
        const float* b1 = bp + (size_t)(k0 + 2 * hi + 1) * N;
        v2f bA, bB, bC, bD;
        bA[0] = b0[0];  bA[1] = b1[0];
        bB[0] = b0[16]; bB[1] = b1[16];
        bC[0] = b0[32]; bC[1] = b1[32];
        bD[0] = b0[48]; bD[1] = b1[48];
        acc0 = __builtin_amdgcn_wmma_f32_16x16x4_f32(false, a, false, bA, (short)0, acc0, false, false);
        acc1 = __builtin_amdgcn_wmma_f32_16x16x4_f32(false, a, false, bB, (short)0, acc1, false, false);
        acc2 = __builtin_amdgcn_wmma_f32_16x16x4_f32(false, a, false, bC, (short)0, acc2, false, false);
        acc3 = __builtin_amdgcn_wmma_f32_16x16x4_f32(false, a, false, bD, (short)0, acc3, false, false);
    }

    auto epi = [&](const v8f& acc, int nsub) {
        const int col = n0 + nsub * 16 + lm;
        const float bv = bias[col];
        #pragma unroll
        for (int r = 0; r < 8; ++r) {
            const int row = m0 + r + 8 * hi;  // C/D layout: VGPR r -> M = r (+8)
            float v = acc[r] + bv;
            if (RESID) v += R[(size_t)row * N + col];
            if (GELU)  v = gelu_exact(v);
            C[(size_t)row * N + col] = v;
        }
    };
    epi(acc0, 0); epi(acc1, 1); epi(acc2, 2); epi(acc3, 3);
}

// ---------------------------------------------------------------------------
// LayerNorm over rows of 128; one wave per row, 8 rows per 256-thread block.
// ---------------------------------------------------------------------------
__global__ __launch_bounds__(256) void ln_kernel(
    const float* __restrict__ X, const float* __restrict__ S,
    const float* __restrict__ Bb, float* __restrict__ Y, int M) {
    const int wave = threadIdx.x >> 5;
    const int lane = threadIdx.x & 31;
    const int row  = blockIdx.x * 8 + wave;
    if (row >= M) return;
    float4 v = ((const float4*)(X + (size_t)row * Dm))[lane];
    float mean = wave_sum32(v.x + v.y + v.z + v.w) * (1.0f / 128.0f);
    float dx = v.x - mean, dy = v.y - mean, dz = v.z - mean, dw = v.w - mean;
    float var = wave_sum32(dx * dx + dy * dy + dz * dz + dw * dw) * (1.0f / 128.0f);
    float inv = rsqrtf(var + 1e-5f);
    float4 sc = ((const float4*)S)[lane];
    float4 bb = ((const float4*)Bb)[lane];
    float4 o;
    o.x = dx * inv * sc.x + bb.x;
    o.y = dy * inv * sc.y + bb.y;
    o.z = dz * inv * sc.z + bb.z;
    o.w = dw * inv * sc.w + bb.w;
    ((float4*)(Y + (size_t)row * Dm))[lane] = o;
}

// ---------------------------------------------------------------------------
// Attention for one (b, head): S = QK^T/4 (WMMA), softmax, AO = P@V (WMMA).
// Q/K/V staged in LDS padded to 64 rows; 4 waves, one 16-row tile each.
// ---------------------------------------------------------------------------
__global__ __launch_bounds__(128) void attn_kernel(
    const float* __restrict__ qkv, float* __restrict__ ao) {
    __shared__ float Qs[64][16];
    __shared__ float Ks[64][16];
    __shared__ float Vs[64][16];
    __shared__ float Sm[64][68];   // 64x64 scores, padded stride

    const int bh = blockIdx.x;
    const int b  = bh / Hn;
    const int hh = bh % Hn;
    const float* base = qkv + (size_t)b * Tlen * (3 * Dm);

    for (int idx = threadIdx.x; idx < 64 * 16; idx += 128) {
        const int r = idx >> 4, c = idx & 15;
        float qv = 0.f, kv = 0.f, vv = 0.f;
        if (r < Tlen) {
            const float* rp = base + (size_t)r * (3 * Dm) + hh * HDm;
            qv = rp[c];
            kv = rp[Dm + c];
            vv = rp[2 * Dm + c];
        }
        Qs[r][c] = qv; Ks[r][c] = kv; Vs[r][c] = vv;
    }
    __syncthreads();

    const int wave = threadIdx.x >> 5;
    const int lane = threadIdx.x & 31;
    const int lm   = lane & 15;
    const int hi   = lane >> 4;
    const int m0   = wave * 16;

    // Phase 1: scores tile row, scale by 1/sqrt(16)=0.25
    #pragma unroll
    for (int nt = 0; nt < 4; ++nt) {
        const int n0 = nt * 16;
        v8f acc = {};
        #pragma unroll
        for (int k0 = 0; k0 < HDm; k0 += 4) {
            v2f a, bb;
            a[0]  = Qs[m0 + lm][k0 + 2 * hi];
            a[1]  = Qs[m0 + lm][k0 + 2 * hi + 1];
            bb[0] = Ks[n0 + lm][k0 + 2 * hi];      // B = K^T
            bb[1] = Ks[n0 + lm][k0 + 2 * hi + 1];
            acc = __builtin_amdgcn_wmma_f32_16x16x4_f32(
                false, a, false, bb, (short)0, acc, false, false);
        }
        #pragma unroll
        for (int r = 0; r < 8; ++r)
            Sm[m0 + r + 8 * hi][n0 + lm] = acc[r] * 0.25f;
    }
    __syncthreads();

    // Softmax over valid 60 columns; zero padded columns
    if (threadIdx.x < Tlen) {
        float* row = Sm[threadIdx.x];
        float mx = -3.4e38f;
        for (int c = 0; c < Tlen; ++c) mx = fmaxf(mx, row[c]);
        float sum = 0.f;
        for (int c = 0; c < Tlen; ++c) { float e = __expf(row[c] - mx); row[c] = e; sum += e; }
        const float invs = 1.0f / sum;
        for (int c = 0; c < Tlen; ++c) row[c] *= invs;
        for (int c = Tlen; c < 64; ++c) row[c] = 0.f;
    }
    __syncthreads();

    // Phase 2: AO = P @ V  (M=16 tile rows per wave, N=16, K=64)
    v8f acc = {};
    #pragma unroll
    for (int k0 = 0; k0 < 64; k0 += 4) {
        v2f a, bb;
        a[0]  = Sm[m0 + lm][k0 + 2 * hi];
        a[1]  = Sm[m0 + lm][k0 + 2 * hi + 1];
        bb[0] = Vs[k0 + 2 * hi][lm];
        bb[1] = Vs[k0 + 2 * hi + 1][lm];
        acc = __builtin_amdgcn_wmma_f32_16x16x4_f32(
            false, a, false, bb, (short)0, acc, false, false);
    }
    #pragma unroll
    for (int r = 0; r < 8; ++r) {
        const int row = m0 + r + 8 * hi;
        if (row < Tlen)
            ao[((size_t)b * Tlen + row) * Dm + hh * HDm + lm] = acc[r];
    }
}

// ---------------------------------------------------------------------------
// Retrieval: per-query top-7 over 1M hist rows (L2-resident 84MB stream),
// softmax(sim/0.1) blend, project through w_ret.
// ---------------------------------------------------------------------------
__device__ __forceinline__ void topk_insert(float* ts, int* ti, float v, int id) {
    if (v <= ts[Ktop - 1]) return;
    int p = Ktop - 1;
    while (p > 0 && ts[p - 1] < v) { ts[p] = ts[p - 1]; ti[p] = ti[p - 1]; --p; }
    ts[p] = v; ti[p] = id;
}

__global__ __launch_bounds__(256) void retrieve_kernel(
    const float* __restrict__ x, const float* __restrict__ hist,
    const float* __restrict__ w_ret, const float* __restrict__ b_ret,
    float* __restrict__ retr) {
    __shared__ float qsh[DIN];
    __shared__ float cs[256 * Ktop];
    __shared__ int   ci[256 * Ktop];
    __shared__ float cs2[32 * Ktop];
    __shared__ int   ci2[32 * Ktop];
    __shared__ float wsh[Ktop];
    __shared__ int   ish[Ktop];
    __shared__ float rsh[DIN];

    const int b = blockIdx.x, tid = threadIdx.x;
    if (tid < DIN) qsh[tid] = x[((size_t)b * Tlen + (Tlen - 1)) * DIN + tid];
    __syncthreads();

    float qn[DIN];
    float nrm = 0.f;
    #pragma unroll
    for (int j = 0; j < DIN; ++j) nrm += qsh[j] * qsh[j];
    const float qi = 1.0f / fmaxf(sqrtf(nrm), 1e-8f);
    #pragma unroll
    for (int j = 0; j < DIN; ++j) qn[j] = qsh[j] * qi;

    float ts[Ktop]; int ti[Ktop];
    #pragma unroll
    for (int k = 0; k < Ktop; ++k) { ts[k] = -3.4e38f; ti[k] = 0; }

    for (int i = tid; i < NHIST; i += 256) {
        const float* hp = hist + (size_t)i * DIN;
        float dot = 0.f, hn = 0.f;
        #pragma unroll
        for (int j = 0; j < DIN; ++j) { const float v = hp[j]; dot += qn[j] * v; hn += v * v; }
        const float sim = dot / fmaxf(sqrtf(hn), 1e-8f);
        topk_insert(ts, ti, sim, i);
    }
    #pragma unroll
    for (int k = 0; k < Ktop; ++k) { cs[tid * Ktop + k] = ts[k]; ci[tid * Ktop + k] = ti[k]; }
    __syncthreads();

    if (tid < 32) {                       // merge 8 candidate lists each
        float bs[Ktop]; int bi[Ktop];
        #pragma unroll
        for (int k = 0; k < Ktop; ++k) { bs[k] = -3.4e38f; bi[k] = 0; }
        for (int s = 0; s < 8; ++s)
            for (int k = 0; k < Ktop; ++k) {
                const int src = (tid * 8 + s) * Ktop + k;
                topk_insert(bs, bi, cs[src], ci[src]);
            }
        #pragma unroll
        for (int k = 0; k < Ktop; ++k) { cs2[tid * Ktop + k] = bs[k]; ci2[tid * Ktop + k] = bi[k]; }
    }
    __syncthreads();

    if (tid == 0) {                       // final merge of 32 lists + softmax
        float fs[Ktop]; int fi[Ktop];
        #pragma unroll
        for (int k = 0; k < Ktop; ++k) { fs[k] = -3.4e38f; fi[k] = 0; }
        for (int s = 0; s < 32 * Ktop; ++s) topk_insert(fs, fi, cs2[s], ci2[s]);
        const float mx = fs[0];           // sorted descending
        float w[Ktop], sum = 0.f;
        #pragma unroll
        for (int k = 0; k < Ktop; ++k) { w[k] = __expf((fs[k] - mx) * 10.0f); sum += w[k]; }
        const float inv = 1.0f / sum;
        #pragma unroll
        for (int k = 0; k < Ktop; ++k) { wsh[k] = w[k] * inv; ish[k] = fi[k]; }
    }
    __syncthreads();

    if (tid < DIN) {                      // retrieved = sum_k w_k * hist[idx_k]
        float r = 0.f;
        #pragma unroll
        for (int k = 0; k < Ktop; ++k) r += wsh[k] * hist[(size_t)ish[k] * DIN + tid];
        rsh[tid] = r;
    }
    __syncthreads();

    if (tid < Dm) {                       // project: retrieved @ w_ret + b_ret
        float acc = b_ret[tid];
        #pragma unroll
        for (int j = 0; j < DIN; ++j) acc += rsh[j] * w_ret[j * Dm + tid];
        retr[(size_t)b * Dm + tid] = acc;
    }
}

// ---------------------------------------------------------------------------
// Final head: out[b] = gelu([h_last, retr] @ wf1 + bf1) @ wf2 + bf2
// ---------------------------------------------------------------------------
__global__ __launch_bounds__(128) void head_kernel(
    const float* __restrict__ h, const float* __restrict__ retr,
    const float* __restrict__ wf1, const float* __restrict__ bf1,
    const float* __restrict__ wf2, const float* __restrict__ bf2,
    float* __restrict__ out) {
    __shared__ float fused[2 * Dm];
    __shared__ float red[Dm];
    const int b = blockIdx.x, tid = threadIdx.x;
    fused[tid]      = h[((size_t)b * Tlen + (Tlen - 1)) * Dm + tid];
    fused[Dm + tid] = retr[(size_t)b * Dm + tid];
    __syncthreads();
    float z = bf1[tid];
    for (int i = 0; i < 2 * Dm; ++i) z += fused[i] * wf1[i * Dm + tid];
    red[tid] = gelu_exact(z) * wf2[tid];
    __syncthreads();
    #pragma unroll
    for (int s = 64; s > 0; s >>= 1) {
        if (tid < s) red[tid] += red[tid + s];
        __syncthreads();
    }
    if (tid == 0) out[b] = red[0] + bf2[0];
}

// ---------------------------------------------------------------------------
extern "C" void kernel_launch(void* const* d_in, const int* in_sizes, int n_in,
                              void* d_out, int out_size, void* d_ws, size_t ws_size,
                              hipStream_t stream) {
    const float* x     = (const float*)d_in[0];
    const float* hist  = (const float*)d_in[1];
    const float* w_in  = (const float*)d_in[2];
    const float* b_in  = (const float*)d_in[3];
    const float* wqkv  = (const float*)d_in[4];
    const float* bqkv  = (const float*)d_in[5];
    const float* wo    = (const float*)d_in[6];
    const float* bo    = (const float*)d_in[7];
    const float* ln1s  = (const float*)d_in[8];
    const float* ln1b  = (const float*)d_in[9];
    const float* w1    = (const float*)d_in[10];
    const float* b1    = (const float*)d_in[11];
    const float* w2    = (const float*)d_in[12];
    const float* b2    = (const float*)d_in[13];
    const float* ln2s  = (const float*)d_in[14];
    const float* ln2b  = (const float*)d_in[15];
    const float* w_ret = (const float*)d_in[16];
    const float* b_ret = (const float*)d_in[17];
    const float* wf1   = (const float*)d_in[18];
    const float* bf1   = (const float*)d_in[19];
    const float* wf2   = (const float*)d_in[20];
    const float* bf2   = (const float*)d_in[21];
    float* out = (float*)d_out;

    const int M = Bsz * Tlen;            // 15360 rows
    float* h    = (float*)d_ws;          // [M, 128]
    float* tmp  = h    + (size_t)M * Dm; // [M, 128]
    float* qkvb = tmp  + (size_t)M * Dm; // [M, 384]
    float* aob  = qkvb + (size_t)M * 3 * Dm; // [M, 128]
    float* ffb  = aob  + (size_t)M * Dm; // [M, 512]
    float* retr = ffb  + (size_t)M * FFm;// [B, 128]

    inproj_kernel<<<M, 128, 0, stream>>>(x, w_in, b_in, h);

    for (int l = 0; l < Lnum; ++l) {
        // QKV projection: [M,128] @ [128,384]
        gemm16x64_f32<false, false><<<dim3(3 * Dm / 64, M / 16), 32, 0, stream>>>(
            h, wqkv + (size_t)l * Dm * 3 * Dm, bqkv + (size_t)l * 3 * Dm,
            nullptr, qkvb, M, 3 * Dm, Dm);
        // Attention per (b, head)
        attn_kernel<<<Bsz * Hn, 128, 0, stream>>>(qkvb, aob);
        // Output projection + residual
        gemm16x64_f32<false, true><<<dim3(Dm / 64, M / 16), 32, 0, stream>>>(
            aob, wo + (size_t)l * Dm * Dm, bo + (size_t)l * Dm,
            h, tmp, M, Dm, Dm);
        ln_kernel<<<M / 8, 256, 0, stream>>>(tmp, ln1s + (size_t)l * Dm, ln1b + (size_t)l * Dm, h, M);
        // FF1 + GELU: [M,128] @ [128,512]
        gemm16x64_f32<true, false><<<dim3(FFm / 64, M / 16), 32, 0, stream>>>(
            h, w1 + (size_t)l * Dm * FFm, b1 + (size_t)l * FFm,
            nullptr, ffb, M, FFm, Dm);
        // FF2 + residual: [M,512] @ [512,128]
        gemm16x64_f32<false, true><<<dim3(Dm / 64, M / 16), 32, 0, stream>>>(
            ffb, w2 + (size_t)l * FFm * Dm, b2 + (size_t)l * Dm,
            h, tmp, M, Dm, FFm);
        ln_kernel<<<M / 8, 256, 0, stream>>>(tmp, ln2s + (size_t)l * Dm, ln2b + (size_t)l * Dm, h, M);
    }

    retrieve_kernel<<<Bsz, 256, 0, stream>>>(x, hist, w_ret, b_ret, retr);
    head_kernel<<<Bsz, 128, 0, stream>>>(h, retr, wf1, bf1, wf2, bf2, out);
}